// CHBertAttentionHead_70076686401674
// MI455X (gfx1250) — compile-verified
//
#include <hip/hip_runtime.h>
#include <hip/hip_bf16.h>
#include <math.h>

// ---------------- problem constants (fixed by reference) ----------------
#define B_   16
#define S_   4096
#define D_   64
#define H_   4
#define NB_  64          // buckets per hash
#define NBIN 256         // H_*NB_ total bucket ids
#define E_   16384       // H_*S_ elements per batch
#define NCH  256         // chunks (= H_*NB_)
#define L_   64          // queries per chunk
#define KL_  128         // keys per chunk (look-one-back)

typedef float v2f __attribute__((ext_vector_type(2)));
typedef float v8f __attribute__((ext_vector_type(8)));

__device__ __forceinline__ v8f wmma_f32(v2f a, v2f b, v8f c) {
  return __builtin_amdgcn_wmma_f32_16x16x4_f32(false, a, false, b, (short)0, c,
                                               false, false);
}

// ---------------- K1: LSH bucket ids ----------------
__global__ void k_buckets(const float* __restrict__ qk,
                          const float* __restrict__ rot,
                          int* __restrict__ buckets) {
  int idx = blockIdx.x * 256 + threadIdx.x;     // [0, B_*H_*S_)
  int b   = idx >> 14;
  int rem = idx & (E_ - 1);
  int h   = rem >> 12;
  int t   = rem & (S_ - 1);
  const float* q = qk + ((size_t)b * S_ + t) * D_;
  float r[32];
#pragma unroll
  for (int i = 0; i < 32; i++) r[i] = 0.f;
  for (int f = 0; f < D_; f++) {
    float qf = q[f];
    const float* rp = rot + f * (H_ * 32) + h * 32;
#pragma unroll
    for (int i = 0; i < 32; i++) r[i] += qf * rp[i];
  }
  float best = r[0];
  int arg = 0;
#pragma unroll
  for (int i = 1; i < 32; i++)
    if (r[i] > best) { best = r[i]; arg = i; }
#pragma unroll
  for (int i = 0; i < 32; i++)
    if (-r[i] > best) { best = -r[i]; arg = 32 + i; }
  buckets[(size_t)b * E_ + rem] = h * NB_ + arg;
}

// ---------------- K2: per-block 256-bin histogram ----------------
__global__ void k_hist(const int* __restrict__ buckets, int* __restrict__ hist) {
  __shared__ int lh[NBIN];
  int b = blockIdx.x >> 6, blk = blockIdx.x & 63;
  lh[threadIdx.x] = 0;
  __syncthreads();
  int bin = buckets[(size_t)b * E_ + blk * 256 + threadIdx.x];
  atomicAdd(&lh[bin], 1);
  __syncthreads();
  // bin-major layout: hist[b][bin][blk]
  hist[(size_t)b * E_ + threadIdx.x * 64 + blk] = lh[threadIdx.x];
}

// ---------------- K3: bin-major exclusive scan (in place) ----------------
__global__ void k_scan(int* __restrict__ hist) {
  __shared__ int tot[NBIN];
  __shared__ int sc[NBIN];
  int b = blockIdx.x, bin = threadIdx.x;
  int* hp = hist + (size_t)b * E_ + bin * 64;
  int s = 0;
  for (int k = 0; k < 64; k++) s += hp[k];
  tot[bin] = s;
  sc[bin] = s;
  __syncthreads();
  for (int off = 1; off < NBIN; off <<= 1) {
    int v = (bin >= off) ? sc[bin - off] : 0;
    __syncthreads();
    sc[bin] += v;
    __syncthreads();
  }
  int run = sc[bin] - tot[bin];   // exclusive bin start
  for (int k = 0; k < 64; k++) { int c = hp[k]; hp[k] = run; run += c; }
}

// ---------------- K4: stable scatter ----------------
__global__ void k_scatter(const int* __restrict__ buckets,
                          const int* __restrict__ hist,
                          int* __restrict__ st, int* __restrict__ undo) {
  __shared__ int lbin[256];
  int b = blockIdx.x >> 6, blk = blockIdx.x & 63;
  int j = blk * 256 + threadIdx.x;
  int bin = buckets[(size_t)b * E_ + j];
  lbin[threadIdx.x] = bin;
  __syncthreads();
  int rank = 0;
  for (int i = 0; i < threadIdx.x; i++) rank += (lbin[i] == bin);
  int pos = hist[(size_t)b * E_ + bin * 64 + blk] + rank;
  st[(size_t)b * E_ + pos] = j & (S_ - 1);   // token index
  undo[(size_t)b * E_ + j] = pos;
}

// ---------------- K5: chunked attention with f32 WMMA ----------------
__global__ void __launch_bounds__(128)
k_attn(const float* __restrict__ qk, const float* __restrict__ v,
       const int* __restrict__ st, float* __restrict__ so,
       float* __restrict__ slog) {
  __shared__ float lq[L_][68];     // queries   64x64  (+4 pad)
  __shared__ float lk[KL_][68];    // keys      128x64 (normalized)
  __shared__ float lv[KL_][68];    // values    128x64
  __shared__ float lp[L_][132];    // exp(dots) 64x128 (+4 pad)
  __shared__ int   qt[L_];
  __shared__ int   kt[KL_];
  __shared__ float lrcp[L_];

  int b = blockIdx.x >> 8, c = blockIdx.x & (NCH - 1);
  int cprev = (c + NCH - 1) & (NCH - 1);
  int tid = threadIdx.x;
  const int* stb = st + (size_t)b * E_;

  // ---- stage Q ----
  for (int i = tid; i < L_ * 16; i += 128) {      // 16 float4 per row
    int row = i >> 4, q4 = i & 15;
    int t = stb[c * L_ + row];
    float4 x = ((const float4*)(qk + ((size_t)b * S_ + t) * D_))[q4];
    *(float4*)&lq[row][q4 * 4] = x;
    if (q4 == 0) qt[row] = t;
  }
  // ---- stage K, V ----
  for (int i = tid; i < KL_ * 16; i += 128) {
    int row = i >> 4, q4 = i & 15;
    int p = (row < L_) ? c * L_ + row : cprev * L_ + (row - L_);
    int t = stb[p];
    *(float4*)&lk[row][q4 * 4] = ((const float4*)(qk + ((size_t)b * S_ + t) * D_))[q4];
    *(float4*)&lv[row][q4 * 4] = ((const float4*)(v  + ((size_t)b * S_ + t) * D_))[q4];
    if (q4 == 0) kt[row] = t;
  }
  __syncthreads();

  // ---- normalize keys ----
  for (int row = tid; row < KL_; row += 128) {
    float ss = 0.f;
    for (int d = 0; d < D_; d++) { float x = lk[row][d]; ss += x * x; }
    float scl = 1.0f / fmaxf(sqrtf(ss), 1e-12f);
    for (int d = 0; d < D_; d++) lk[row][d] *= scl;
  }
  __syncthreads();

  int wave = tid >> 5, lane = tid & 31;
  int m = lane & 15, hi = lane >> 4;
  int rbase = wave * 16;

  // A fragments (Q strip 16x64), ISA layout: lane half splits K {0,1}/{2,3}
  v2f afq[16];
#pragma unroll
  for (int k4 = 0; k4 < 16; k4++) {
    afq[k4].x = lq[rbase + m][4 * k4 + 2 * hi];
    afq[k4].y = lq[rbase + m][4 * k4 + 2 * hi + 1];
  }

  // ---- dots = Q @ K^T * D^-0.5, masked ----
  for (int nt = 0; nt < 8; nt++) {
    v8f acc = {0.f, 0.f, 0.f, 0.f, 0.f, 0.f, 0.f, 0.f};
#pragma unroll
    for (int k4 = 0; k4 < 16; k4++) {
      v2f bf;
      bf.x = lk[nt * 16 + m][4 * k4 + 2 * hi];
      bf.y = lk[nt * 16 + m][4 * k4 + 2 * hi + 1];
      acc = wmma_f32(afq[k4], bf, acc);
    }
    int col = nt * 16 + m;
    int ktc = kt[col];
#pragma unroll
    for (int e = 0; e < 8; e++) {
      int row = rbase + e + 8 * hi;
      float val = acc[e] * 0.125f;           // 64^-0.5
      if (qt[row] == ktc) val = -5e4f;       // self-attn mask
      lp[row][col] = val;
    }
  }

  // ---- row softmax (wave-local rows; LDS ops in-order per wave) ----
  {
    int r = rbase + m;
    int c0 = hi * 64;
    float mx = -3.402823466e38f;
    for (int cc = 0; cc < 64; cc++) mx = fmaxf(mx, lp[r][c0 + cc]);
    mx = fmaxf(mx, __shfl_xor(mx, 16, 32));
    float ssum = 0.f;
    for (int cc = 0; cc < 64; cc++) {
      float e = __expf(lp[r][c0 + cc] - mx);
      lp[r][c0 + cc] = e;
      ssum += e;
    }
    ssum += __shfl_xor(ssum, 16, 32);
    float lse = mx + __logf(ssum);
    if (hi == 0) {
      lrcp[r] = 1.0f / ssum;
      slog[(size_t)b * E_ + c * L_ + r] = lse;
    }
  }

  // ---- bo = P @ V ----
  for (int nt2 = 0; nt2 < 4; nt2++) {
    v8f acc = {0.f, 0.f, 0.f, 0.f, 0.f, 0.f, 0.f, 0.f};
#pragma unroll
    for (int k4 = 0; k4 < 32; k4++) {
      v2f af, bf;
      af.x = lp[rbase + m][4 * k4 + 2 * hi];
      af.y = lp[rbase + m][4 * k4 + 2 * hi + 1];
      bf.x = lv[4 * k4 + 2 * hi][nt2 * 16 + m];
      bf.y = lv[4 * k4 + 2 * hi + 1][nt2 * 16 + m];
      acc = wmma_f32(af, bf, acc);
    }
#pragma unroll
    for (int e = 0; e < 8; e++) {
      int row = rbase + e + 8 * hi;
      size_t idx = ((size_t)b * E_ + c * L_ + row) * D_ + nt2 * 16 + m;
      so[idx] = acc[e] * lrcp[row];
    }
  }
}

// ---------------- K6: unsort + combine hash rounds ----------------
__global__ void k_combine(const float* __restrict__ so,
                          const float* __restrict__ slog,
                          const int* __restrict__ undo,
                          float* __restrict__ out) {
  int b = blockIdx.x >> 12, t = blockIdx.x & (S_ - 1);
  int d = threadIdx.x;
  const int* ub = undo + (size_t)b * E_;
  int p[H_];
  float lg[H_];
#pragma unroll
  for (int h = 0; h < H_; h++) {
    p[h] = ub[h * S_ + t];
    lg[h] = slog[(size_t)b * E_ + p[h]];
  }
  float mx = fmaxf(fmaxf(lg[0], lg[1]), fmaxf(lg[2], lg[3]));
  float w[H_], ws = 0.f;
#pragma unroll
  for (int h = 0; h < H_; h++) { w[h] = __expf(lg[h] - mx); ws += w[h]; }
  float acc = 0.f;
#pragma unroll
  for (int h = 0; h < H_; h++)
    acc += w[h] * so[((size_t)b * E_ + p[h]) * D_ + d];
  out[((size_t)b * S_ + t) * D_ + d] = acc / ws;
}

// ---------------- launcher ----------------
extern "C" void kernel_launch(void* const* d_in, const int* in_sizes, int n_in,
                              void* d_out, int out_size, void* d_ws, size_t ws_size,
                              hipStream_t stream) {
  const float* qk  = (const float*)d_in[0];
  const float* v   = (const float*)d_in[1];
  const float* rot = (const float*)d_in[2];
  float* out = (float*)d_out;

  char* ws = (char*)d_ws;
  size_t off = 0;
  int*   buckets = (int*)(ws + off);  off += (size_t)B_ * E_ * sizeof(int);
  int*   hist    = (int*)(ws + off);  off += (size_t)B_ * E_ * sizeof(int);
  int*   st      = (int*)(ws + off);  off += (size_t)B_ * E_ * sizeof(int);
  int*   undo    = (int*)(ws + off);  off += (size_t)B_ * E_ * sizeof(int);
  float* slog    = (float*)(ws + off); off += (size_t)B_ * E_ * sizeof(float);
  float* so      = (float*)(ws + off); off += (size_t)B_ * E_ * D_ * sizeof(float);

  k_buckets<<<(B_ * H_ * S_) / 256, 256, 0, stream>>>(qk, rot, buckets);
  k_hist   <<<B_ * 64, 256, 0, stream>>>(buckets, hist);
  k_scan   <<<B_, 256, 0, stream>>>(hist);
  k_scatter<<<B_ * 64, 256, 0, stream>>>(buckets, hist, st, undo);
  k_attn   <<<B_ * NCH, 128, 0, stream>>>(qk, v, st, so, slog);
  k_combine<<<B_ * S_, 64, 0, stream>>>(so, slog, undo, out);
}